// GNN_12283606467757
// MI455X (gfx1250) — compile-verified
//
#include <hip/hip_runtime.h>
#include <hip/hip_bf16.h>
#include <math.h>

// Problem constants (match reference)
#define Nn    4096
#define Ee    131072
#define Ff    64
#define HIDd  64
#define OUTd  86
#define Bb    16
#define HEADSh 4
#define DH    16

typedef float v2f __attribute__((ext_vector_type(2)));
typedef float v8f __attribute__((ext_vector_type(8)));

// D = A(16x4 f32) * B(4x16 f32) + C  -- native CDNA5 f32 WMMA
__device__ __forceinline__ v8f wmma4(v2f a, v2f b, v8f c) {
  return __builtin_amdgcn_wmma_f32_16x16x4_f32(false, a, false, b, (short)0, c,
                                               false, false);
}

// ---------------- degree / norm ----------------
__global__ void k_initdeg(float* deg) {
  int i = blockIdx.x * blockDim.x + threadIdx.x;
  if (i < Nn) deg[i] = 1.0f;  // self-loop weight 1
}

__global__ void k_degedges(const int* __restrict__ ei,
                           const float* __restrict__ ea, float* deg) {
  int e = blockIdx.x * blockDim.x + threadIdx.x;
  if (e < Ee) atomicAdd(&deg[ei[Ee + e]], ea[e]);
}

__global__ void k_dis(const float* __restrict__ deg, float* dis) {
  int i = blockIdx.x * blockDim.x + threadIdx.x;
  if (i < Nn) {
    float d = deg[i];
    dis[i] = d > 0.0f ? rsqrtf(fmaxf(d, 1e-12f)) : 0.0f;
  }
}

__global__ void k_norm(const int* __restrict__ ei, const float* __restrict__ ea,
                       const float* __restrict__ dis, float* normv) {
  int e = blockIdx.x * blockDim.x + threadIdx.x;
  if (e < Ee) {
    normv[e] = dis[ei[e]] * ea[e] * dis[ei[Ee + e]];
  } else if (e < Ee + Nn) {
    int i = e - Ee;
    normv[e] = dis[i] * dis[i];
  }
}

// ---------------- WMMA GEMM: C[M,NC] = A[M,K] @ W[NC,K]^T (+bias) ----------------
// One wave per 16x16 tile. A rows stride K, W rows stride K, C rows stride NC.
// K is compile-time (64 everywhere here) so the k-loop fully unrolls; the ragged
// NC tail (NC=86) is handled branchlessly via a clamped row + cndmask-zeroed frag.
template <int K>
__global__ void k_gemm_wt(const float* __restrict__ A, const float* __restrict__ W,
                          const float* __restrict__ bias, float* __restrict__ Cout,
                          int NC, int totTiles, int tilesN) {
  int wave = blockIdx.x * (blockDim.x >> 5) + (threadIdx.x >> 5);
  if (wave >= totTiles) return;
  int lane = threadIdx.x & 31;
  int m = lane & 15;
  int hi = lane >> 4;
  int rb = wave / tilesN;
  int cb = wave - rb * tilesN;

  int arow = rb * 16 + m;             // A-frag row (M)
  int wrow = cb * 16 + m;             // B-frag column = W row (N)
  bool wok = wrow < NC;
  int wrowc = wok ? wrow : (NC - 1);  // clamp: load stays in-bounds, branchless
  const float* Ar = A + (size_t)arow * K;
  const float* Wr = W + (size_t)wrowc * K;

  v8f acc = {};
#pragma unroll
  for (int ks = 0; ks < K / 4; ++ks) {
    int k0 = ks * 4 + hi * 2;
    v2f a = *(const v2f*)(Ar + k0);
    v2f b = *(const v2f*)(Wr + k0);
    if (!wok) { b.x = 0.0f; b.y = 0.0f; }   // v_cndmask, no EXEC round-trip
    acc = wmma4(a, b, acc);
  }

  int ccol = cb * 16 + m;
  float badd = (bias != nullptr && wok) ? bias[ccol] : 0.0f;
#pragma unroll
  for (int r = 0; r < 8; ++r) {
    if (wok) {
      int orow = rb * 16 + r + hi * 8;
      Cout[(size_t)orow * NC + ccol] = acc[r] + badd;
    }
  }
}

// ---------------- bias fill / scatter / relu ----------------
__global__ void k_fillbias(float* __restrict__ agg, const float* __restrict__ b,
                           int Fd, int total) {
  int tid = blockIdx.x * blockDim.x + threadIdx.x;
  if (tid < total) agg[tid] = b[tid % Fd];
}

__global__ void k_scatter(const float* __restrict__ hsrc, float* __restrict__ agg,
                          const int* __restrict__ ei,
                          const float* __restrict__ normv, int Fd, int total) {
  int tid = blockIdx.x * blockDim.x + threadIdx.x;
  if (tid >= total) return;
  int e = tid / Fd;
  int f = tid - e * Fd;
  int s, d;
  if (e < Ee) { s = ei[e]; d = ei[Ee + e]; }
  else        { s = d = e - Ee; }
  atomicAdd(&agg[(size_t)d * Fd + f], normv[e] * hsrc[(size_t)s * Fd + f]);
}

__global__ void k_relu(float* x, int total) {
  int tid = blockIdx.x * blockDim.x + threadIdx.x;
  if (tid < total) x[tid] = fmaxf(x[tid], 0.0f);
}

// ---------------- flash attention, one wave per (head, 16-row block) ----------------
__global__ void __launch_bounds__(32) k_attn(const float* __restrict__ qkv,
                                             float* __restrict__ O) {
  __shared__ float pl[256];  // 16x16 score tile staging (C-layout -> A-layout)
  int wave = blockIdx.x;
  int h = wave >> 8;      // head
  int rb = wave & 255;    // query row-block
  int lane = threadIdx.x & 31;
  int m = lane & 15;
  int hi = lane >> 4;
  const int STR = 3 * HIDd;  // qkv row stride (192)

  // Q A-frags, pre-scaled by 1/sqrt(dh) = 0.25
  v2f qf[4];
  const float* qrow = qkv + (size_t)(rb * 16 + m) * STR + h * DH;
#pragma unroll
  for (int ks = 0; ks < 4; ++ks) {
    v2f t = *(const v2f*)(qrow + ks * 4 + hi * 2);
    qf[ks].x = t.x * 0.25f;
    qf[ks].y = t.y * 0.25f;
  }

  v8f acc = {};
  float mrow = -INFINITY;  // running row max (A-layout row = m, dup across halves)
  float lrow = 0.0f;       // running row denom

  for (int jb = 0; jb < Nn / 16; ++jb) {
    // S tile = Q @ K^T  (B-frag: K[j][d], d contiguous)
    const float* krow = qkv + (size_t)(jb * 16 + m) * STR + HIDd + h * DH;
    v8f s = {};
#pragma unroll
    for (int ks = 0; ks < 4; ++ks) {
      v2f kf = *(const v2f*)(krow + ks * 4 + hi * 2);
      s = wmma4(qf[ks], kf, s);
    }

    // C-layout -> LDS -> A-layout
#pragma unroll
    for (int r = 0; r < 8; ++r) pl[(r + hi * 8) * 16 + m] = s[r];
    __syncthreads();
    v2f pf[4];
#pragma unroll
    for (int ks = 0; ks < 4; ++ks)
      pf[ks] = *(const v2f*)(&pl[m * 16 + ks * 4 + hi * 2]);
    __syncthreads();

    // online softmax in A-layout: lane holds 8 of row m's 16 values
    float lm = fmaxf(fmaxf(fmaxf(pf[0].x, pf[0].y), fmaxf(pf[1].x, pf[1].y)),
                     fmaxf(fmaxf(pf[2].x, pf[2].y), fmaxf(pf[3].x, pf[3].y)));
    lm = fmaxf(lm, __shfl_xor(lm, 16, 32));
    float mnew = fmaxf(mrow, lm);
    float c = __expf(mrow - mnew);  // 0 on first tile (exp(-inf))
    float ls = 0.0f;
#pragma unroll
    for (int ks = 0; ks < 4; ++ks) {
      pf[ks].x = __expf(pf[ks].x - mnew);
      pf[ks].y = __expf(pf[ks].y - mnew);
      ls += pf[ks].x + pf[ks].y;
    }
    ls += __shfl_xor(ls, 16, 32);
    lrow = lrow * c + ls;
    mrow = mnew;

    // rescale accumulator (C-layout rows need c of row r+8*hi, held at lane r+8*hi)
#pragma unroll
    for (int r = 0; r < 8; ++r) {
      float cr = __shfl(c, r + hi * 8, 32);
      acc[r] *= cr;
    }

    // acc += P @ V  (B-frag: V[j][d], lane = d, VGPR/half = j)
#pragma unroll
    for (int ks = 0; ks < 4; ++ks) {
      int k0 = ks * 4 + hi * 2;
      v2f vf;
      vf.x = qkv[(size_t)(jb * 16 + k0 + 0) * STR + 2 * HIDd + h * DH + m];
      vf.y = qkv[(size_t)(jb * 16 + k0 + 1) * STR + 2 * HIDd + h * DH + m];
      acc = wmma4(pf[ks], vf, acc);
    }
  }

  // normalize and store O[N, HID] (per-head column slice)
#pragma unroll
  for (int r = 0; r < 8; ++r) {
    float lr = __shfl(lrow, r + hi * 8, 32);
    O[(size_t)(rb * 16 + r + hi * 8) * HIDd + h * DH + m] = acc[r] / lr;
  }
}

// ---------------- pooling ----------------
__global__ void k_zero(float* out, float* cnt) {
  int i = blockIdx.x * blockDim.x + threadIdx.x;
  if (i < Bb * OUTd) out[i] = 0.0f;
  if (i < Bb) cnt[i] = 0.0f;
}

__global__ void k_pool(const float* __restrict__ h2, const int* __restrict__ batch,
                       float* out, float* cnt) {
  int tid = blockIdx.x * blockDim.x + threadIdx.x;
  if (tid >= Nn * OUTd) return;
  int n = tid / OUTd;
  int f = tid - n * OUTd;
  int b = batch[n];
  atomicAdd(&out[b * OUTd + f], h2[tid]);
  if (f == 0) atomicAdd(&cnt[b], 1.0f);
}

__global__ void k_pooldiv(float* out, const float* __restrict__ cnt) {
  int tid = blockIdx.x * blockDim.x + threadIdx.x;
  if (tid < Bb * OUTd) out[tid] /= fmaxf(cnt[tid / OUTd], 1.0f);
}

static inline int cdiv(int a, int b) { return (a + b - 1) / b; }

extern "C" void kernel_launch(void* const* d_in, const int* in_sizes, int n_in,
                              void* d_out, int out_size, void* d_ws, size_t ws_size,
                              hipStream_t stream) {
  const float* x    = (const float*)d_in[0];
  const int*   ei   = (const int*)d_in[1];
  const float* ea   = (const float*)d_in[2];
  const int*   bat  = (const int*)d_in[3];
  const float* W1   = (const float*)d_in[4];
  const float* b1   = (const float*)d_in[5];
  const float* Win  = (const float*)d_in[6];
  const float* bin  = (const float*)d_in[7];
  const float* Wout = (const float*)d_in[8];
  const float* bout = (const float*)d_in[9];
  const float* W2   = (const float*)d_in[10];
  const float* b2   = (const float*)d_in[11];
  float* out = (float*)d_out;

  // workspace layout (floats, all offsets 16B-aligned)
  float* ws    = (float*)d_ws;
  float* deg   = ws;                    // 4096
  float* dis   = deg + Nn;              // 4096
  float* normv = dis + Nn;              // 135168
  float* h0    = normv + (Ee + Nn);     // 262144  x @ W1^T
  float* h1    = h0 + Nn * HIDd;        // 262144  relu(agg1 + b1)
  float* qkv   = h1 + Nn * HIDd;        // 786432
  float* o     = qkv + Nn * 3 * HIDd;   // 262144  attention out
  float* oproj = o + Nn * HIDd;         // 262144  o @ Wout^T + b_out
  float* h2pre = oproj + Nn * HIDd;     // 352256  oproj @ W2^T
  float* agg2  = h2pre + Nn * OUTd;     // 352256
  float* cnt   = agg2 + Nn * OUTd;      // 16

  const int T = 256;

  // degree / normalization coefficients
  k_initdeg<<<cdiv(Nn, T), T, 0, stream>>>(deg);
  k_degedges<<<cdiv(Ee, T), T, 0, stream>>>(ei, ea, deg);
  k_dis<<<cdiv(Nn, T), T, 0, stream>>>(deg, dis);
  k_norm<<<cdiv(Ee + Nn, T), T, 0, stream>>>(ei, ea, dis, normv);

  // GCN1: h0 = x @ W1^T  (M=4096, K=64, NC=64)
  {
    int tilesN = HIDd / 16, tot = (Nn / 16) * tilesN;
    k_gemm_wt<Ff><<<cdiv(tot, 8), T, 0, stream>>>(x, W1, nullptr, h0, HIDd, tot, tilesN);
  }
  // agg1 = b1; scatter norm*h0[src] -> agg1[dst]; relu
  k_fillbias<<<cdiv(Nn * HIDd, T), T, 0, stream>>>(h1, b1, HIDd, Nn * HIDd);
  k_scatter<<<cdiv((Ee + Nn) * HIDd, T), T, 0, stream>>>(h0, h1, ei, normv, HIDd,
                                                         (Ee + Nn) * HIDd);
  k_relu<<<cdiv(Nn * HIDd, T), T, 0, stream>>>(h1, Nn * HIDd);

  // QKV projection: qkv = h1 @ Win^T + b_in  (NC=192)
  {
    int tilesN = (3 * HIDd) / 16, tot = (Nn / 16) * tilesN;
    k_gemm_wt<HIDd><<<cdiv(tot, 8), T, 0, stream>>>(h1, Win, bin, qkv, 3 * HIDd, tot, tilesN);
  }

  // flash attention: HEADS * (N/16) waves, one wave per block
  k_attn<<<HEADSh * (Nn / 16), 32, 0, stream>>>(qkv, o);

  // output projection: oproj = o @ Wout^T + b_out
  {
    int tilesN = HIDd / 16, tot = (Nn / 16) * tilesN;
    k_gemm_wt<HIDd><<<cdiv(tot, 8), T, 0, stream>>>(o, Wout, bout, oproj, HIDd, tot, tilesN);
  }

  // GCN2: h2pre = oproj @ W2^T  (NC=86, ragged tail tile masked branchlessly)
  {
    int tilesN = cdiv(OUTd, 16), tot = (Nn / 16) * tilesN;
    k_gemm_wt<HIDd><<<cdiv(tot, 8), T, 0, stream>>>(oproj, W2, nullptr, h2pre, OUTd, tot, tilesN);
  }
  k_fillbias<<<cdiv(Nn * OUTd, T), T, 0, stream>>>(agg2, b2, OUTd, Nn * OUTd);
  k_scatter<<<cdiv((Ee + Nn) * OUTd, T), T, 0, stream>>>(h2pre, agg2, ei, normv, OUTd,
                                                         (Ee + Nn) * OUTd);

  // global mean pool
  k_zero<<<cdiv(Bb * OUTd, T), T, 0, stream>>>(out, cnt);
  k_pool<<<cdiv(Nn * OUTd, T), T, 0, stream>>>(agg2, bat, out, cnt);
  k_pooldiv<<<cdiv(Bb * OUTd, T), T, 0, stream>>>(out, cnt);
}